// Orthogonal_86526411145717
// MI455X (gfx1250) — compile-verified
//
#include <hip/hip_runtime.h>
#include <math.h>
#include <stdint.h>

typedef __attribute__((ext_vector_type(2))) float v2f;
typedef __attribute__((ext_vector_type(8))) float v8f;

#define DIM   1792
#define BATCH 8192
#define DT    (DIM/16)    // 112 tiles per dimension
#define PROW  72          // padded LDS panel row stride (floats): 2-row half-wave offset
                          // = 16 banks -> conflict-free ds_load_b32

// ---------------------------------------------------------------------------
// K1: Vn = U / ||U_row||    (one block per row)
// ---------------------------------------------------------------------------
__global__ void k_normalize(const float* __restrict__ U, float* __restrict__ Vn) {
    const int row = blockIdx.x;
    __shared__ float red[256];
    float s = 0.f;
    for (int c = threadIdx.x; c < DIM; c += 256) {
        float u = U[(size_t)row * DIM + c];
        s += u * u;
    }
    red[threadIdx.x] = s;
    __syncthreads();
    for (int off = 128; off > 0; off >>= 1) {
        if (threadIdx.x < off) red[threadIdx.x] += red[threadIdx.x + off];
        __syncthreads();
    }
    const float inv = 1.0f / sqrtf(red[0]);
    for (int c = threadIdx.x; c < DIM; c += 256)
        Vn[(size_t)row * DIM + c] = U[(size_t)row * DIM + c] * inv;
}

// ---------------------------------------------------------------------------
// K2: G = Vn * Vn^T, lower triangle (incl. diagonal tiles) only.
// One wave per 16x16 tile; fp32 WMMA 16x16x4.
// ---------------------------------------------------------------------------
__global__ void k_gram_lower(const float* __restrict__ Vn, float* __restrict__ G) {
    const int ti = blockIdx.x, tj = blockIdx.y;
    if (tj > ti) return;                       // uniform per-wave exit: EXEC stays all-1
    const int lane = threadIdx.x;
    const int half = lane >> 4, l16 = lane & 15;
    const size_t arow = (size_t)(ti * 16 + l16) * DIM;
    const size_t brow = (size_t)(tj * 16 + l16) * DIM;
    v8f c = {};
    for (int k = 0; k < DIM; k += 4) {
        const int kk = k + 2 * half;
        v2f a, b;
        a[0] = Vn[arow + kk];     a[1] = Vn[arow + kk + 1];
        b[0] = Vn[brow + kk];     b[1] = Vn[brow + kk + 1];
        c = __builtin_amdgcn_wmma_f32_16x16x4_f32(false, a, false, b, (short)0, c, false, false);
    }
#pragma unroll
    for (int r = 0; r < 8; ++r) {
        const int m = ti * 16 + r + 8 * half;
        G[(size_t)m * DIM + tj * 16 + l16] = c[r];
    }
}

// ---------------------------------------------------------------------------
// K3: one block-row of the forward substitution
//   R  = 2*Vn[kb] - 2 * G[kb, 0:kb*16) @ Wt[0:kb*16, :]        (WMMA GEMM)
//   Wt[kb*16+r] = R[r] - 2*sum_{s<r} G[kb*16+r, kb*16+s]*Wt[kb*16+s]   (in-LDS)
// ---------------------------------------------------------------------------
__global__ void k_solve_blockrow(const float* __restrict__ Vn, const float* __restrict__ G,
                                 float* __restrict__ Wt, int kb) {
    const int tn = blockIdx.x;
    const int lane = threadIdx.x;
    const int half = lane >> 4, l16 = lane & 15;
    __shared__ float R[16][17];

    const size_t arow = (size_t)(kb * 16 + l16) * DIM;
    const int    ncol = tn * 16 + l16;
    v8f acc = {};
    for (int k = 0; k < kb * 16; k += 4) {
        const int kk = k + 2 * half;
        v2f a, b;
        a[0] = G[arow + kk];             a[1] = G[arow + kk + 1];
        b[0] = Wt[(size_t)kk * DIM + ncol];
        b[1] = Wt[(size_t)(kk + 1) * DIM + ncol];
        acc = __builtin_amdgcn_wmma_f32_16x16x4_f32(false, a, false, b, (short)0, acc, false, false);
    }
#pragma unroll
    for (int r = 0; r < 8; ++r) {
        const int row = r + 8 * half;
        R[row][l16] = 2.0f * Vn[(size_t)(kb * 16 + row) * DIM + tn * 16 + l16] - 2.0f * acc[r];
    }
    __syncthreads();
    if (lane < 16) {                        // lane owns one column; serial over 16 rows
        for (int r = 1; r < 16; ++r) {
            float s = R[r][lane];
            const float* gr = &G[(size_t)(kb * 16 + r) * DIM + kb * 16];
            for (int ss = 0; ss < r; ++ss)
                s -= 2.0f * gr[ss] * R[ss][lane];
            R[r][lane] = s;
        }
    }
    __syncthreads();
    if (lane < 16) {
        for (int r = 0; r < 16; ++r)
            Wt[(size_t)(kb * 16 + r) * DIM + tn * 16 + lane] = R[r][lane];
    }
}

// ---------------------------------------------------------------------------
// Async panel stage: each of the 256 threads issues one 16B async copy of the
// 16x64 fp32 B panel (rows k0..k0+15, cols n0..n0+63) into LDS (ASYNCcnt path).
// ---------------------------------------------------------------------------
__device__ __forceinline__ void stage_panel_async(const float* __restrict__ gbase,
                                                  int n0, int k0,
                                                  float (*panel)[PROW]) {
    const int t  = threadIdx.x;
    const int r  = t >> 4;        // panel row 0..15
    const int ch = t & 15;        // 16B chunk 0..15 within row
    uint32_t lds_off = (uint32_t)(uintptr_t)(&panel[r][ch * 4]);
    uint32_t goff    = (uint32_t)(((uint32_t)(k0 + r) * (uint32_t)BATCH +
                                   (uint32_t)n0 + (uint32_t)ch * 4u) * 4u);
    asm volatile("global_load_async_to_lds_b128 %0, %1, %2"
                 :: "v"(lds_off), "v"(goff), "s"(gbase) : "memory");
}

// ---------------------------------------------------------------------------
// Paneled GEMM core: 8 waves/block, wave w computes rows (bx*8+w)*16..+15,
// cols n0..n0+63.  B panel double-buffered in LDS via async loads.
//   FINAL=0:  D = A @ Bsrc                    (Y = Vn @ X)
//   FINAL=1:  D = Xref - A^T-indexed @ Bsrc   (out = X - Wt^T @ Y)
// ---------------------------------------------------------------------------
template <int FINAL>
__global__ void k_panel_gemm(const float* __restrict__ A,     // Vn (row-major) or Wt (read as columns)
                             const float* __restrict__ Bsrc,  // X or Y   (DIM x BATCH)
                             const float* __restrict__ Xref,  // X for FINAL epilogue (unused otherwise)
                             float* __restrict__ Dout) {
    __shared__ __align__(16) float panelB[2][16][PROW];

    const int wave = threadIdx.x >> 5;
    const int lane = threadIdx.x & 31;
    const int half = lane >> 4, l16 = lane & 15;
    const int m0   = (blockIdx.x * 8 + wave) * 16;     // wave's M tile base
    const int n0   = blockIdx.y * 64;                  // block's N panel base
    const int nc   = n0 + l16;

    v8f acc0 = {}, acc1 = {}, acc2 = {}, acc3 = {};

    stage_panel_async(Bsrc, n0, 0, panelB[0]);
    int buf = 0;
    for (int k0 = 0; k0 < DIM; k0 += 16) {
        const bool has_next = (k0 + 16) < DIM;
        if (has_next) {
            stage_panel_async(Bsrc, n0, k0 + 16, panelB[buf ^ 1]);
            asm volatile("s_wait_asynccnt 0x1" ::: "memory");   // current panel done
        } else {
            asm volatile("s_wait_asynccnt 0x0" ::: "memory");
        }
        __syncthreads();                                        // all waves' chunks visible

#pragma unroll
        for (int ks = 0; ks < 16; ks += 4) {
            const int kk = ks + 2 * half;       // panel row for this half-wave
            const int kg = k0 + kk;             // global k
            v2f a;
            if (FINAL) {                        // A = Wt^T : column reads, coalesced over lanes
                a[0] = A[(size_t)kg * DIM + m0 + l16];
                a[1] = A[(size_t)(kg + 1) * DIM + m0 + l16];
            } else {                            // A = Vn row-major
                a[0] = A[(size_t)(m0 + l16) * DIM + kg];
                a[1] = A[(size_t)(m0 + l16) * DIM + kg + 1];
            }
            v2f b0, b1, b2, b3;
            b0[0] = panelB[buf][kk][l16 +  0];  b0[1] = panelB[buf][kk + 1][l16 +  0];
            b1[0] = panelB[buf][kk][l16 + 16];  b1[1] = panelB[buf][kk + 1][l16 + 16];
            b2[0] = panelB[buf][kk][l16 + 32];  b2[1] = panelB[buf][kk + 1][l16 + 32];
            b3[0] = panelB[buf][kk][l16 + 48];  b3[1] = panelB[buf][kk + 1][l16 + 48];
            acc0 = __builtin_amdgcn_wmma_f32_16x16x4_f32(false, a, false, b0, (short)0, acc0, false, false);
            acc1 = __builtin_amdgcn_wmma_f32_16x16x4_f32(false, a, false, b1, (short)0, acc1, false, false);
            acc2 = __builtin_amdgcn_wmma_f32_16x16x4_f32(false, a, false, b2, (short)0, acc2, false, false);
            acc3 = __builtin_amdgcn_wmma_f32_16x16x4_f32(false, a, false, b3, (short)0, acc3, false, false);
        }
        __syncthreads();                        // buf may be overwritten next iteration
        buf ^= 1;
    }

#pragma unroll
    for (int r = 0; r < 8; ++r) {
        const int m = m0 + r + 8 * half;
        const size_t base = (size_t)m * BATCH + n0 + l16;
        if (FINAL) {
            Dout[base +  0] = Xref[base +  0] - acc0[r];
            Dout[base + 16] = Xref[base + 16] - acc1[r];
            Dout[base + 32] = Xref[base + 32] - acc2[r];
            Dout[base + 48] = Xref[base + 48] - acc3[r];
        } else {
            Dout[base +  0] = acc0[r];
            Dout[base + 16] = acc1[r];
            Dout[base + 32] = acc2[r];
            Dout[base + 48] = acc3[r];
        }
    }
}

// ---------------------------------------------------------------------------
// Host launcher. Inputs: d_in[0] = X (D x B fp32), d_in[1] = U (D x D fp32).
// Workspace layout (floats): Vn | G | Wt | Y
// ---------------------------------------------------------------------------
extern "C" void kernel_launch(void* const* d_in, const int* in_sizes, int n_in,
                              void* d_out, int out_size, void* d_ws, size_t ws_size,
                              hipStream_t stream) {
    (void)in_sizes; (void)n_in; (void)out_size; (void)ws_size;
    const float* X = (const float*)d_in[0];
    const float* U = (const float*)d_in[1];
    float* out = (float*)d_out;
    float* ws  = (float*)d_ws;

    float* Vn = ws;
    float* G  = ws + (size_t)DIM * DIM;
    float* Wt = ws + 2 * (size_t)DIM * DIM;
    float* Y  = ws + 3 * (size_t)DIM * DIM;

    // 1) normalize reflector rows
    k_normalize<<<DIM, 256, 0, stream>>>(U, Vn);

    // 2) Gram matrix, lower triangle
    dim3 gG(DT, DT);
    k_gram_lower<<<gG, 32, 0, stream>>>(Vn, G);

    // 3) blocked unit-lower triangular solve (sequential over 112 block-rows)
    for (int kb = 0; kb < DT; ++kb)
        k_solve_blockrow<<<DT, 32, 0, stream>>>(Vn, G, Wt, kb);

    // 4) Y = Vn @ X         (async-staged B panels, 8 waves/block)
    dim3 gB(DT / 8, BATCH / 64);
    k_panel_gemm<0><<<gB, 256, 0, stream>>>(Vn, X, X, Y);

    // 5) out = X - Wt^T @ Y
    k_panel_gemm<1><<<gB, 256, 0, stream>>>(Wt, Y, X, out);
}